// AttentionModel_88338887344423
// MI455X (gfx1250) — compile-verified
//
#include <hip/hip_runtime.h>
#include <stdint.h>

typedef float v2f __attribute__((ext_vector_type(2)));
typedef float v8f __attribute__((ext_vector_type(8)));

#define Bt 32
#define St 2048
#define Ht 1024
#define KC 128      // K-chunk per double-buffer step
#define RS 130      // padded LDS row stride (floats) -> avoids bank conflicts

// ---------------- Kernel A: partial column reductions of W_attn with v ----------------
// t[c] = sum_h v[h] * W[h, c]  for c in [0, 2048). whv = t[0:1024], u = t[1024:2048].
__global__ __launch_bounds__(256) void wcol_partial(const float* __restrict__ W,
                                                    const float* __restrict__ v,
                                                    float* __restrict__ part) {
    const int c  = blockIdx.x * 256 + threadIdx.x;   // 0..2047
    const int h0 = blockIdx.y * 64;                  // 16 h-segments
    float acc = 0.f;
    for (int h = h0; h < h0 + 64; ++h)
        acc = fmaf(v[h], W[(size_t)h * 2048 + c], acc);
    part[blockIdx.y * 2048 + c] = acc;
}

__global__ __launch_bounds__(256) void wcol_reduce(const float* __restrict__ part,
                                                   float* __restrict__ whv,
                                                   float* __restrict__ u) {
    const int c = blockIdx.x * 256 + threadIdx.x;    // 0..2047
    float acc = 0.f;
    #pragma unroll
    for (int p = 0; p < 16; ++p) acc += part[p * 2048 + c];
    if (c < 1024) whv[c] = acc; else u[c - 1024] = acc;
}

// ---------------- Kernel C: c_b = hidden[b]·whv + b_attn·v ----------------
__global__ __launch_bounds__(256) void cb_kernel(const float* __restrict__ hid,
                                                 const float* __restrict__ ba,
                                                 const float* __restrict__ v,
                                                 const float* __restrict__ whv,
                                                 float* __restrict__ cb) {
    __shared__ float red[256];
    const int b = blockIdx.x, tid = threadIdx.x;
    float acc = 0.f;
    for (int d = tid; d < Ht; d += 256)
        acc += hid[(size_t)b * Ht + d] * whv[d] + ba[d] * v[d];
    red[tid] = acc; __syncthreads();
    for (int off = 128; off > 0; off >>= 1) {
        if (tid < off) red[tid] += red[tid + off];
        __syncthreads();
    }
    if (tid == 0) cb[b] = red[0];
}

// ---------------- Kernel D: energy[b,s] = enc[b,s,:]·u + c_b ----------------
// 8 waves/block, 16 encoder rows per wave, K double-buffered through LDS via
// global_load_async_to_lds_b128; dot via V_WMMA_F32_16X16X4_F32.
__global__ __launch_bounds__(256) void energy_kernel(const float* __restrict__ enc,
                                                     const float* __restrict__ u,
                                                     const float* __restrict__ cb,
                                                     float* __restrict__ out) {
    __shared__ float su[Ht];
    __shared__ float tiles[8][2][16 * RS];           // 8 waves x double buffer x 16x130
    const int tid = threadIdx.x, lane = tid & 31, wave = tid >> 5;
    const int b  = blockIdx.x;
    const int s0 = blockIdx.y * 128 + wave * 16;     // 16 rows per wave

    for (int i = tid; i < Ht; i += 256) su[i] = u[i];
    __syncthreads();

    const float* gbase = enc + ((size_t)b * St + (size_t)s0) * Ht;
    const int m16 = lane & 15;       // A-matrix row handled by this lane
    const int hs  = lane >> 4;       // K-half select (lanes 16..31 hold K+2,K+3)

    auto issue = [&](int buf, int k0) {
        unsigned lbase = (unsigned)(uintptr_t)(&tiles[wave][buf][0]) + (unsigned)lane * 16u;
        unsigned long long ga = (unsigned long long)(uintptr_t)(gbase + k0)
                              + (unsigned long long)lane * 16u;
        #pragma unroll
        for (int m = 0; m < 16; ++m) {   // one 512B row-chunk per instruction
            asm volatile("global_load_async_to_lds_b128 %0, %1, off th:TH_LOAD_NT"
                         :: "v"(lbase), "v"(ga) : "memory");
            lbase += (unsigned)(RS * 4);
            ga    += (unsigned long long)(Ht * 4);
        }
    };

    v8f c0 = {0.f,0.f,0.f,0.f,0.f,0.f,0.f,0.f};
    v8f c1 = {0.f,0.f,0.f,0.f,0.f,0.f,0.f,0.f};

    issue(0, 0);
    for (int step = 0; step < 8; ++step) {
        const int k0 = step * KC;
        if (step < 7) {
            issue((step + 1) & 1, k0 + KC);
            asm volatile("s_wait_asynccnt 16" ::: "memory");  // current buffer landed
        } else {
            asm volatile("s_wait_asynccnt 0" ::: "memory");
        }
        const float* tb   = &tiles[wave][step & 1][0];
        const float* arow = tb + m16 * RS + 2 * hs;   // A: K = kk+2*hs, kk+2*hs+1
        const float* brow = su + k0 + 2 * hs;         // B: u chunk, broadcast across N
        #pragma unroll 4
        for (int kk = 0; kk < KC; kk += 8) {
            v2f a0 = *(const v2f*)(arow + kk);
            v2f b0 = *(const v2f*)(brow + kk);
            c0 = __builtin_amdgcn_wmma_f32_16x16x4_f32(false, a0, false, b0,
                                                       (short)0, c0, false, false);
            v2f a1 = *(const v2f*)(arow + kk + 4);
            v2f b1 = *(const v2f*)(brow + kk + 4);
            c1 = __builtin_amdgcn_wmma_f32_16x16x4_f32(false, a1, false, b1,
                                                       (short)0, c1, false, false);
        }
    }

    // D layout: VGPR j, lanes 0-15 hold M=j, lanes 16-31 hold M=j+8 (all N equal).
    if (m16 == 0) {
        const float base = cb[b];
        float* o = out + (size_t)b * St + s0 + 8 * hs;
        #pragma unroll
        for (int j = 0; j < 8; ++j) o[j] = c0[j] + c1[j] + base;
    }
}

// ---------------- Kernel E: in-place softmax over s (2048) per batch ----------------
__global__ __launch_bounds__(256) void softmax_kernel(float* __restrict__ eo) {
    __shared__ float red[256];
    const int b = blockIdx.x, tid = threadIdx.x;
    float e[8];
    float m = -3.0e38f;
    #pragma unroll
    for (int i = 0; i < 8; ++i) {
        e[i] = eo[(size_t)b * St + i * 256 + tid];
        m = fmaxf(m, e[i]);
    }
    red[tid] = m; __syncthreads();
    for (int off = 128; off > 0; off >>= 1) {
        if (tid < off) red[tid] = fmaxf(red[tid], red[tid + off]);
        __syncthreads();
    }
    m = red[0]; __syncthreads();
    float s = 0.f;
    #pragma unroll
    for (int i = 0; i < 8; ++i) { e[i] = expf(e[i] - m); s += e[i]; }
    red[tid] = s; __syncthreads();
    for (int off = 128; off > 0; off >>= 1) {
        if (tid < off) red[tid] += red[tid + off];
        __syncthreads();
    }
    const float inv = 1.0f / red[0];
    #pragma unroll
    for (int i = 0; i < 8; ++i)
        eo[(size_t)b * St + i * 256 + tid] = e[i] * inv;
}

extern "C" void kernel_launch(void* const* d_in, const int* in_sizes, int n_in,
                              void* d_out, int out_size, void* d_ws, size_t ws_size,
                              hipStream_t stream) {
    const float* enc = (const float*)d_in[0];
    const float* hid = (const float*)d_in[1];
    const float* W   = (const float*)d_in[2];
    const float* ba  = (const float*)d_in[3];
    const float* v   = (const float*)d_in[4];
    float* out = (float*)d_out;
    float* ws  = (float*)d_ws;

    float* u    = ws;            // 1024
    float* whv  = ws + 1024;     // 1024
    float* cb   = ws + 2048;     // 32
    float* part = ws + 4096;     // 16*2048

    wcol_partial <<<dim3(8, 16), 256, 0, stream>>>(W, v, part);
    wcol_reduce  <<<8,           256, 0, stream>>>(part, whv, u);
    cb_kernel    <<<Bt,          256, 0, stream>>>(hid, ba, v, whv, cb);
    energy_kernel<<<dim3(Bt, 16),256, 0, stream>>>(enc, u, cb, out);
    softmax_kernel<<<Bt,         256, 0, stream>>>(out);
}